// CustomAttn_74981539054040
// MI455X (gfx1250) — compile-verified
//
#include <hip/hip_runtime.h>
#include <math.h>

// Problem constants (match reference)
#define BATCH 8
#define CIN   8
#define LEN   8192
#define CH    512
#define KW    5
#define W23ROWS 16  // w23 padded from 8 rows to 16 (WMMA M dim), pad rows = 0

typedef __attribute__((ext_vector_type(2))) float v2f;
typedef __attribute__((ext_vector_type(8))) float v8f;

// D = A(16x4) * B(4x16) + C(16x16), fp32 WMMA.
// 8-arg form: (neg_a, A, neg_b, B, c_mod, C, reuse_a, reuse_b)
__device__ __forceinline__ v8f wmma4(v2f a, v2f b, v8f c) {
  return __builtin_amdgcn_wmma_f32_16x16x4_f32(false, a, false, b, (short)0, c,
                                               false, false);
}

// 8-byte aligned pair load -> single global_load_b64
__device__ __forceinline__ v2f ld2(const float* p) {
  return *(const v2f*)p;
}

// Lane decomposition per ISA 7.12.2 (wave32, 32-bit fragments):
//  lm = lane & 15, half = lane >> 4, kk = 2*half
//  A frag elem i -> A[m0+lm][k0+kk+i]
//  B frag elem i -> B[k0+kk+i][n0+lm]
//  D vgpr r      -> D[m0+r+8*half][n0+lm]

// ---------------------------------------------------------------------------
// Kernel 1: h = relu(conv1d(x, w1, b1, pad=2))   [B, CH, LEN]
// 5 shifted 1x1 GEMMs: M=CH, K=CIN, N=LEN. One wave per 16x16 tile.
// Boundary handling: clamp address + mask multiply (no EXEC divergence).
// ---------------------------------------------------------------------------
__global__ __launch_bounds__(32) void conv1_relu_wmma(
    const float* __restrict__ x, const float* __restrict__ w1,
    const float* __restrict__ b1, float* __restrict__ h) {
  const int lane = threadIdx.x & 31;
  const int lm = lane & 15, half = lane >> 4, kk = half << 1;
  const int l0 = blockIdx.x * 16;
  const int m0 = blockIdx.y * 16;
  const int b  = blockIdx.z;
  const float* xb = x + (size_t)b * CIN * LEN;
  v8f acc = {};
#pragma unroll
  for (int k = 0; k < KW; ++k) {
    const int l = l0 + lm + k - 2;
    const float msk = (l >= 0 && l < LEN) ? 1.0f : 0.0f;
    const int lc = min(max(l, 0), LEN - 1);
    for (int c0 = 0; c0 < CIN; c0 += 4) {
      v2f a, bf;
#pragma unroll
      for (int i = 0; i < 2; ++i) {
        const int ci = c0 + kk + i;
        a[i]  = w1[(m0 + lm) * (CIN * KW) + ci * KW + k];
        bf[i] = msk * xb[(size_t)ci * LEN + lc];
      }
      acc = wmma4(a, bf, acc);
    }
  }
  float* hb = h + (size_t)b * CH * LEN;
#pragma unroll
  for (int r = 0; r < 8; ++r) {
    const int m = m0 + r + (half << 3);
    const float v = acc[r] + b1[m];
    hb[(size_t)m * LEN + l0 + lm] = v > 0.0f ? v : 0.0f;
  }
}

// ---------------------------------------------------------------------------
// Kernel 2: G = H * H^T  per batch  [CH, CH], K = LEN.
// One wave computes a 32x32 tile (2x2 register blocking of 16x16 WMMA tiles).
// All fragment loads are 8B-aligned b64 row reads of h.
// ---------------------------------------------------------------------------
__global__ __launch_bounds__(32) void gram_wmma(const float* __restrict__ h,
                                                float* __restrict__ G) {
  const int lane = threadIdx.x & 31;
  const int lm = lane & 15, half = lane >> 4, kk = half << 1;
  const int m0 = (blockIdx.x >> 4) * 32;
  const int n0 = (blockIdx.x & 15) * 32;
  const int b = blockIdx.y;
  const float* hb = h + (size_t)b * CH * LEN;
  const float* pa0 = hb + (size_t)(m0 + lm) * LEN + kk;
  const float* pa1 = hb + (size_t)(m0 + 16 + lm) * LEN + kk;
  const float* pb0 = hb + (size_t)(n0 + lm) * LEN + kk;
  const float* pb1 = hb + (size_t)(n0 + 16 + lm) * LEN + kk;
  v8f a00 = {}, a01 = {}, a10 = {}, a11 = {};
  for (int l0 = 0; l0 < LEN; l0 += 4) {
    v2f fa0 = ld2(pa0 + l0);
    v2f fa1 = ld2(pa1 + l0);
    v2f fb0 = ld2(pb0 + l0);
    v2f fb1 = ld2(pb1 + l0);
    a00 = wmma4(fa0, fb0, a00);
    a01 = wmma4(fa0, fb1, a01);
    a10 = wmma4(fa1, fb0, a10);
    a11 = wmma4(fa1, fb1, a11);
  }
  float* Gb = G + (size_t)b * CH * CH;
#pragma unroll
  for (int r = 0; r < 8; ++r) {
    const int mA = m0 + r + (half << 3);
    const int mB = m0 + 16 + r + (half << 3);
    Gb[(size_t)mA * CH + n0 + lm]      = a00[r];
    Gb[(size_t)mA * CH + n0 + 16 + lm] = a01[r];
    Gb[(size_t)mB * CH + n0 + lm]      = a10[r];
    Gb[(size_t)mB * CH + n0 + 16 + lm] = a11[r];
  }
}

// ---------------------------------------------------------------------------
// Kernel 3: batched 512x512x512 GEMM, C = A * op(B). TB=1 -> B transposed.
// Per-batch strides (0 for shared weight matrices). 32x32 tile per wave.
// ---------------------------------------------------------------------------
template <bool TB>
__global__ __launch_bounds__(32) void gemm512_wmma(
    const float* __restrict__ A, long aStride, const float* __restrict__ Bm,
    long bStride, float* __restrict__ Cm, long cStride) {
  const int lane = threadIdx.x & 31;
  const int lm = lane & 15, half = lane >> 4, kk = half << 1;
  const int m0 = (blockIdx.x >> 4) * 32;
  const int n0 = (blockIdx.x & 15) * 32;
  const int b = blockIdx.y;
  A += (size_t)b * aStride;
  Bm += (size_t)b * bStride;
  Cm += (size_t)b * cStride;
  const float* pa0 = A + (size_t)(m0 + lm) * CH + kk;
  const float* pa1 = A + (size_t)(m0 + 16 + lm) * CH + kk;
  v8f a00 = {}, a01 = {}, a10 = {}, a11 = {};
  for (int k0 = 0; k0 < CH; k0 += 4) {
    v2f fa0 = ld2(pa0 + k0);
    v2f fa1 = ld2(pa1 + k0);
    v2f fb0, fb1;
    if (TB) {
      fb0 = ld2(Bm + (size_t)(n0 + lm) * CH + k0 + kk);
      fb1 = ld2(Bm + (size_t)(n0 + 16 + lm) * CH + k0 + kk);
    } else {
#pragma unroll
      for (int i = 0; i < 2; ++i) {
        const int k = k0 + kk + i;
        fb0[i] = Bm[(size_t)k * CH + n0 + lm];
        fb1[i] = Bm[(size_t)k * CH + n0 + 16 + lm];
      }
    }
    a00 = wmma4(fa0, fb0, a00);
    a01 = wmma4(fa0, fb1, a01);
    a10 = wmma4(fa1, fb0, a10);
    a11 = wmma4(fa1, fb1, a11);
  }
#pragma unroll
  for (int r = 0; r < 8; ++r) {
    const int mA = m0 + r + (half << 3);
    const int mB = m0 + 16 + r + (half << 3);
    Cm[(size_t)mA * CH + n0 + lm]      = a00[r];
    Cm[(size_t)mA * CH + n0 + 16 + lm] = a01[r];
    Cm[(size_t)mB * CH + n0 + lm]      = a10[r];
    Cm[(size_t)mB * CH + n0 + 16 + lm] = a11[r];
  }
}

// ---------------------------------------------------------------------------
// Kernel 4: column softmax over axis=-2 of A_ [B, CH, CH].
// Thread t owns column d = t (coalesced across threads), loops c (stride CH).
// ---------------------------------------------------------------------------
__global__ void softmax_dim_c(float* __restrict__ A) {
  const int d = blockIdx.x * blockDim.x + threadIdx.x;
  const int b = blockIdx.y;
  float* Ab = A + (size_t)b * CH * CH + d;
  float mx = -INFINITY;
  for (int c = 0; c < CH; ++c) mx = fmaxf(mx, Ab[(size_t)c * CH]);
  float s = 0.0f;
  for (int c = 0; c < CH; ++c) s += __expf(Ab[(size_t)c * CH] - mx);
  const float inv = 1.0f / s;
  for (int c = 0; c < CH; ++c)
    Ab[(size_t)c * CH] = __expf(Ab[(size_t)c * CH] - mx) * inv;
}

// ---------------------------------------------------------------------------
// Kernel 5: h[:, l0:l0+16] = relu(M @ h[:, l0:l0+16]) in place, per batch.
// Block = 512 threads (16 waves). Stage the 512x16 column tile to LDS, then
// each wave produces 32 output rows (2x 16x16 WMMA tiles, shared B frag).
// ---------------------------------------------------------------------------
__global__ __launch_bounds__(512) void av_inplace_relu(
    const float* __restrict__ M, float* __restrict__ h) {
  __shared__ float tile[CH * 16];  // 32 KB
  const int tid = threadIdx.x;
  const int l0 = blockIdx.x * 16;
  const int b = blockIdx.y;
  float* hb = h + (size_t)b * CH * LEN;
  const float* Mb = M + (size_t)b * CH * CH;
  {
    const int j = tid & 15, cbase = tid >> 4;  // cbase 0..31
#pragma unroll
    for (int r = 0; r < 16; ++r) {
      const int c = cbase + (r << 5);
      tile[c * 16 + j] = hb[(size_t)c * LEN + l0 + j];
    }
  }
  __syncthreads();
  const int lane = tid & 31, lm = lane & 15, half = lane >> 4, kk = half << 1;
  const int m0 = (tid >> 5) * 32, m1 = m0 + 16;
  const float* pa0 = Mb + (size_t)(m0 + lm) * CH + kk;
  const float* pa1 = Mb + (size_t)(m1 + lm) * CH + kk;
  v8f acc0 = {}, acc1 = {};
  for (int k0 = 0; k0 < CH; k0 += 4) {
    v2f a0 = ld2(pa0 + k0);
    v2f a1 = ld2(pa1 + k0);
    v2f bf;
#pragma unroll
    for (int i = 0; i < 2; ++i) bf[i] = tile[(k0 + kk + i) * 16 + lm];
    acc0 = wmma4(a0, bf, acc0);
    acc1 = wmma4(a1, bf, acc1);
  }
#pragma unroll
  for (int r = 0; r < 8; ++r) {
    float v0 = acc0[r], v1 = acc1[r];
    hb[(size_t)(m0 + r + (half << 3)) * LEN + l0 + lm] = v0 > 0.0f ? v0 : 0.0f;
    hb[(size_t)(m1 + r + (half << 3)) * LEN + l0 + lm] = v1 > 0.0f ? v1 : 0.0f;
  }
}

// ---------------------------------------------------------------------------
// Kernel 6: fold conv3 into conv2 with zero-padded output rows:
//   w23p[ci][c][k] = sum_o w3[ci][o] * w2[o][c][k]   (ci < 8; rows 8..15 = 0)
//   bias23[ci]     = b3[ci] + sum_o w3[ci][o] * b2[o]
// ---------------------------------------------------------------------------
__global__ void fuse_w23(const float* __restrict__ w2,
                         const float* __restrict__ w3,
                         const float* __restrict__ b2,
                         const float* __restrict__ b3,
                         float* __restrict__ w23p, float* __restrict__ bias23) {
  const int idx = blockIdx.x * blockDim.x + threadIdx.x;
  if (idx < W23ROWS * CH * KW) {
    const int ci = idx / (CH * KW);
    const int rem = idx % (CH * KW);
    float s = 0.0f;
    if (ci < CIN) {
      for (int o = 0; o < CH; ++o)
        s += w3[ci * CH + o] * w2[(size_t)o * CH * KW + rem];
    }
    w23p[idx] = s;
  }
  if (idx < CIN) {
    float s = b3[idx];
    for (int o = 0; o < CH; ++o) s += w3[idx * CH + o] * b2[o];
    bias23[idx] = s;
  }
}

// ---------------------------------------------------------------------------
// Kernel 7: out = conv1d(y, w23p, bias23, pad=2).  M=16 (rows 8..15 are zero),
// K=CH per tap, 5 taps. One wave per 16-wide L tile.
// Boundary handling: clamp address + mask multiply (no EXEC divergence).
// ---------------------------------------------------------------------------
__global__ __launch_bounds__(32) void conv23_wmma(
    const float* __restrict__ y, const float* __restrict__ w23p,
    const float* __restrict__ bias23, float* __restrict__ out) {
  const int lane = threadIdx.x & 31;
  const int lm = lane & 15, half = lane >> 4, kk = half << 1;
  const int l0 = blockIdx.x * 16;
  const int b = blockIdx.y;
  const float* yb = y + (size_t)b * CH * LEN;
  v8f acc = {};
#pragma unroll
  for (int k = 0; k < KW; ++k) {
    const int l = l0 + lm + k - 2;
    const float msk = (l >= 0 && l < LEN) ? 1.0f : 0.0f;
    const int lc = min(max(l, 0), LEN - 1);
    for (int c0 = 0; c0 < CH; c0 += 4) {
      v2f a, bf;
#pragma unroll
      for (int i = 0; i < 2; ++i) {
        const int c = c0 + kk + i;
        a[i]  = w23p[lm * (CH * KW) + c * KW + k];
        bf[i] = msk * yb[(size_t)c * LEN + lc];
      }
      acc = wmma4(a, bf, acc);
    }
  }
  if (half == 0) {
    float* ob = out + (size_t)b * CIN * LEN;
#pragma unroll
    for (int r = 0; r < 8; ++r)
      ob[(size_t)r * LEN + l0 + lm] = acc[r] + bias23[r];
  }
}

// ---------------------------------------------------------------------------
extern "C" void kernel_launch(void* const* d_in, const int* in_sizes, int n_in,
                              void* d_out, int out_size, void* d_ws,
                              size_t ws_size, hipStream_t stream) {
  (void)in_sizes; (void)n_in; (void)out_size; (void)ws_size;
  const float* x  = (const float*)d_in[0];
  const float* w1 = (const float*)d_in[1];
  const float* b1 = (const float*)d_in[2];
  const float* wq = (const float*)d_in[3];
  const float* wk = (const float*)d_in[4];
  const float* wv = (const float*)d_in[5];
  const float* w2 = (const float*)d_in[6];
  const float* b2 = (const float*)d_in[7];
  const float* w3 = (const float*)d_in[8];
  const float* b3 = (const float*)d_in[9];
  float* out = (float*)d_out;

  char* ws = (char*)d_ws;
  size_t off = 0;
  float* h    = (float*)(ws + off); off += (size_t)BATCH * CH * LEN * 4;  // 128 MB
  float* G    = (float*)(ws + off); off += (size_t)BATCH * CH * CH * 4;   // 8 MB
  float* T    = (float*)(ws + off); off += (size_t)BATCH * CH * CH * 4;   // 8 MB
  float* Amat = (float*)(ws + off); off += (size_t)BATCH * CH * CH * 4;   // 8 MB
  float* Mm   = (float*)(ws + off); off += (size_t)BATCH * CH * CH * 4;   // 8 MB
  float* w23p = (float*)(ws + off); off += (size_t)W23ROWS * CH * KW * 4;
  float* bias23 = (float*)(ws + off); off += 64;

  const long CC = (long)CH * CH;

  // w23 fusion is independent of the main chain.
  fuse_w23<<<dim3((W23ROWS * CH * KW + 255) / 256), 256, 0, stream>>>(
      w2, w3, b2, b3, w23p, bias23);

  // h = relu(conv1(x))
  conv1_relu_wmma<<<dim3(LEN / 16, CH / 16, BATCH), 32, 0, stream>>>(x, w1, b1, h);

  // G = H H^T (per batch)
  gram_wmma<<<dim3(256, BATCH), 32, 0, stream>>>(h, G);

  // T = Wq * G ;  A_ = T * Wk^T
  gemm512_wmma<false><<<dim3(256, BATCH), 32, 0, stream>>>(wq, 0, G, CC, T, CC);
  gemm512_wmma<true><<<dim3(256, BATCH), 32, 0, stream>>>(T, CC, wk, 0, Amat, CC);

  // softmax over axis=-2 (column-wise over c)
  softmax_dim_c<<<dim3(CH / 256, BATCH), 256, 0, stream>>>(Amat);

  // M = A * Wv
  gemm512_wmma<false><<<dim3(256, BATCH), 32, 0, stream>>>(Amat, CC, wv, 0, Mm, CC);

  // h <- relu(M * h)  (in place per L tile)
  av_inplace_relu<<<dim3(LEN / 16, BATCH), 512, 0, stream>>>(Mm, h);

  // out = conv1d(h, w23p) + bias23
  conv23_wmma<<<dim3(LEN / 16, BATCH), 32, 0, stream>>>(h, w23p, bias23, out);
}